// MoEGraphEncoder_12790412607647
// MI455X (gfx1250) — compile-verified
//
#include <hip/hip_runtime.h>
#include <hip/hip_bf16.h>
#include <math.h>

// MoE constant-curvature graph encoder for gfx1250 (MI455X).
// - fp32 GEMMs on the matrix pipe via V_WMMA_F32_16X16X4_F32 (full precision).
// - Weight matrix staged into LDS by the Tensor Data Mover (tensor_load_to_lds,
//   TENSORcnt-tracked) when the builtin exists; cooperative-copy fallback.
// - Each wave owns a 16-row stripe and computes all NOUT/16 column tiles
//   (8 independent WMMA accumulation chains for deep XDL pipelining).
// - Edge aggregation (the HBM-bound part): wave-per-edge float atomics with
//   global_prefetch lookahead on the gather side.

#define DD   128
#define GH   32
#define BB   128
#define KEXP 4
#define EPSF 1e-15f

// output layout: scaled [B,512] | weights [B,4] | dists [B,4] | tau [4]
#define W_OFF   (BB * 512)            // 65536
#define D_OFF   (W_OFF + BB * 4)      // 66048
#define TAU_OFF (D_OFF + BB * 4)      // 66560

typedef __attribute__((ext_vector_type(2))) float v2f;
typedef __attribute__((ext_vector_type(8))) float v8f;
typedef __attribute__((ext_vector_type(4))) unsigned int uv4;
typedef __attribute__((ext_vector_type(4))) int iv4;
typedef __attribute__((ext_vector_type(8))) int iv8;

#if defined(__has_builtin)
#if __has_builtin(__builtin_amdgcn_tensor_load_to_lds)
#define HAVE_TDM 1
#endif
#endif

// ---------------- device helpers ----------------

__device__ __forceinline__ float wave_sum(float v) {
#pragma unroll
  for (int m = 16; m >= 1; m >>= 1) v += __shfl_xor(v, m, 32);
  return v;
}
__device__ __forceinline__ float dot4(float4 a, float4 b) {
  return a.x * b.x + a.y * b.y + a.z * b.z + a.w * b.w;
}
__device__ __forceinline__ float4 scl4(float4 a, float s) {
  return make_float4(a.x * s, a.y * s, a.z * s, a.w * s);
}
__device__ __forceinline__ float tan_k(float x, float k) {
  if (k > 0.f) { float sk = sqrtf(k);  return tanf(x * sk) / sk; }
  if (k < 0.f) { float sk = sqrtf(-k); return tanhf(x * sk) / sk; }
  return x;
}
__device__ __forceinline__ float artan_k(float x, float k) {
  if (k > 0.f) { float sk = sqrtf(k);  return atanf(x * sk) / sk; }
  if (k < 0.f) {
    float sk = sqrtf(-k);
    float t = fminf(fmaxf(x * sk, -1.f + 1e-7f), 1.f - 1e-7f);
    return atanhf(t) / sk;
  }
  return x;
}
__device__ __forceinline__ float proj_scale(float n, float k) {
  if (k < 0.f) {
    float mx = (1.f - 1e-5f) / sqrtf(-k);
    if (n > mx) return mx / n;
  }
  return 1.f;
}

// ---------------- generic utility kernels ----------------

__global__ void k_fill(float* __restrict__ p, float v, int n) {
  int i = blockIdx.x * blockDim.x + threadIdx.x;
  if (i < n) p[i] = v;
}

__global__ void k_deg_accum(const int* __restrict__ col, float* __restrict__ deg, int E) {
  int i = blockIdx.x * blockDim.x + threadIdx.x;
  if (i < E) atomicAdd(&deg[col[i]], 1.0f);
}

__global__ void k_rsqrt_inplace(float* __restrict__ p, int n) {
  int i = blockIdx.x * blockDim.x + threadIdx.x;
  if (i < n) p[i] = rsqrtf(p[i]);
}

__global__ void k_counts(const int* __restrict__ batch, float* __restrict__ counts, int n) {
  int i = blockIdx.x * blockDim.x + threadIdx.x;
  if (i < n) atomicAdd(&counts[batch[i]], 1.0f);
}

// ---------------- WMMA fp32 GEMM:  out[M][NOUT] = X[M][KD] * W[NOUT][KD]^T ----------------
// Block = 256 threads = 8 waves. W staged in LDS once per block (TDM when
// available); each wave owns one 16-row stripe and computes all NOUT/16 column
// tiles (independent accumulation chains -> deep WMMA pipelining).
template <int KD, int NOUT>
__global__ void k_gemm_lds(const float* __restrict__ X, const float* __restrict__ W,
                           float* __restrict__ out, int M) {
  constexpr int NT = NOUT / 16;
  __shared__ float lw[NOUT * KD];

#if defined(HAVE_TDM)
  // Tensor Data Mover: DMA the 2D row-major tile W[NOUT][KD] (fp32) into LDS.
  if (threadIdx.x < 32) {   // one wave issues; TDM ignores EXEC
    unsigned lds0 = (unsigned)(uintptr_t)&lw[0];
    unsigned long long ga = (unsigned long long)(uintptr_t)W;
    uv4 g0;
    g0.x = 1u;                                              // count=1, user D#
    g0.y = lds0;                                            // lds_addr (bytes)
    g0.z = (unsigned)(ga & 0xFFFFFFFFu);                    // global_addr[31:0]
    g0.w = (unsigned)((ga >> 32) & 0x01FFFFFFu) | 0x80000000u;  // addr[56:32] | type=2
    iv8 g1;
    g1[0] = (int)(2u << 16);                                // data_size = 4 bytes
    g1[1] = (int)(((unsigned)KD & 0xFFFFu) << 16);          // tensor_dim0 lo (bits 79:48)
    g1[2] = (int)((((unsigned)KD >> 16) & 0xFFFFu) |
                  (((unsigned)NOUT & 0xFFFFu) << 16));      // dim0 hi | tensor_dim1 lo
    g1[3] = (int)((((unsigned)NOUT >> 16) & 0xFFFFu) |
                  (((unsigned)KD & 0xFFFFu) << 16));        // dim1 hi | tile_dim0 = KD
    g1[4] = (int)(unsigned)NOUT;                            // tile_dim1 = NOUT, tile_dim2=0
    g1[5] = (int)(unsigned)KD;                              // tensor_dim0_stride = KD
    g1[6] = 0;
    g1[7] = 0;
    iv4 z4 = {0, 0, 0, 0};
#if __clang_major__ >= 23
    iv8 z8 = {0, 0, 0, 0, 0, 0, 0, 0};
    __builtin_amdgcn_tensor_load_to_lds(g0, g1, z4, z4, z8, 0);
#else
    __builtin_amdgcn_tensor_load_to_lds(g0, g1, z4, z4, 0);
#endif
    __builtin_amdgcn_s_wait_tensorcnt(0);
  }
  __syncthreads();
#else
  for (int i = threadIdx.x; i < NOUT * KD / 4; i += blockDim.x)
    ((float4*)lw)[i] = ((const float4*)W)[i];
  __syncthreads();
#endif

  const int wave = threadIdx.x >> 5;      // 0..7
  const int lane = threadIdx.x & 31;
  const int half = lane >> 4;             // K split across half-waves
  const int l15  = lane & 15;
  const int rt   = blockIdx.x * 8 + wave; // 16-row tile index

  int arow = rt * 16 + l15;
  int ar   = arow < M ? arow : (M - 1);   // clamp loads; stores guarded below
  const float* Xr = X + (size_t)ar * KD;

  v8f acc[NT] = {};

#pragma unroll 4
  for (int k0 = 0; k0 < KD; k0 += 4) {
    int koff = k0 + 2 * half;             // lanes 0-15: K=k0,k0+1 ; 16-31: k0+2,k0+3
    float2 a2 = *(const float2*)(Xr + koff);
    v2f a; a.x = a2.x; a.y = a2.y;
#pragma unroll
    for (int t = 0; t < NT; ++t) {
      float2 b2 = *(const float2*)(lw + (t * 16 + l15) * KD + koff);
      v2f b; b.x = b2.x; b.y = b2.y;
      acc[t] = __builtin_amdgcn_wmma_f32_16x16x4_f32(false, a, false, b, (short)0,
                                                     acc[t], false, false);
    }
  }

  // C/D layout: VGPR v -> row = v + 8*half, col = l15
  if (rt * 16 + 16 <= M) {                // full tile: straight-line stores
#pragma unroll
    for (int t = 0; t < NT; ++t) {
      float* Or = out + (size_t)(rt * 16 + half * 8) * NOUT + t * 16 + l15;
#pragma unroll
      for (int v = 0; v < 8; ++v) Or[(size_t)v * NOUT] = acc[t][v];
    }
  } else if (rt * 16 < M) {               // partial tile: guarded stores
#pragma unroll
    for (int t = 0; t < NT; ++t) {
      float* Or = out + (size_t)(rt * 16 + half * 8) * NOUT + t * 16 + l15;
#pragma unroll
      for (int v = 0; v < 8; ++v) {
        int r = rt * 16 + half * 8 + v;
        if (r < M) Or[(size_t)v * NOUT] = acc[t][v];
      }
    }
  }
}

// ---------------- row-wise manifold kernels (one wave32 per 128-wide row) ----------------

__global__ void k_expmap0(const float* __restrict__ src, float* __restrict__ dst,
                          float kc, int nrows) {
  int w = (blockIdx.x * blockDim.x + threadIdx.x) >> 5;
  int lane = threadIdx.x & 31;
  if (w >= nrows) return;
  float4 u = ((const float4*)(src + (size_t)w * DD))[lane];
  float n = sqrtf(wave_sum(dot4(u, u)));
  float nc = fmaxf(n, EPSF);
  float f = tan_k(nc, kc) / nc;
  float on = f * n;
  f *= proj_scale(on, kc);
  ((float4*)(dst + (size_t)w * DD))[lane] = scl4(u, f);
}

// mobius_matvec fixup + mobius_add(bias point) + logmap0 (mx buffer becomes t),
// fused with self-loop aggregation init: agg[n] = dinv[n]^2 * t[n].
__global__ void k_expert_post(const float* __restrict__ xm, float* __restrict__ mx,
                              float* __restrict__ agg, const float* __restrict__ dinv,
                              const float* __restrict__ bias, float kc, int nrows) {
  int w = (blockIdx.x * blockDim.x + threadIdx.x) >> 5;
  int lane = threadIdx.x & 31;
  if (w >= nrows) return;
  float4 x4 = ((const float4*)(xm + (size_t)w * DD))[lane];
  float4 m4 = ((const float4*)(mx + (size_t)w * DD))[lane];

  float xn = sqrtf(wave_sum(dot4(x4, x4)));
  float mn = sqrtf(wave_sum(dot4(m4, m4)));
  float asum = wave_sum(fabsf(m4.x) + fabsf(m4.y) + fabsf(m4.z) + fabsf(m4.w));
  float xnc = fmaxf(xn, EPSF), mnc = fmaxf(mn, EPSF);

  float rf = tan_k(mnc / xnc * artan_k(xnc, kc), kc) / mnc;
  if (asum == 0.f) rf = 0.f;
  float hn = rf * mn;
  float ps = proj_scale(hn, kc);
  rf *= ps; hn *= ps;
  float4 h4 = scl4(m4, rf);

  // bias point = expmap0(bias)
  float4 b4 = ((const float4*)bias)[lane];
  float bn = sqrtf(wave_sum(dot4(b4, b4)));
  float bnc = fmaxf(bn, EPSF);
  float bf = tan_k(bnc, kc) / bnc;
  float yn = bf * bn;
  float ps2 = proj_scale(yn, kc);
  bf *= ps2; yn *= ps2;
  float4 y4 = scl4(b4, bf);

  // mobius_add(h, y)
  float x2 = hn * hn, y2 = yn * yn;
  float xy = wave_sum(dot4(h4, y4));
  float ca = 1.f - 2.f * kc * xy - kc * y2;
  float cb = 1.f + kc * x2;
  float den = fmaxf(1.f - 2.f * kc * xy + kc * kc * x2 * y2, 1e-15f);
  float4 s4 = make_float4((ca * h4.x + cb * y4.x) / den, (ca * h4.y + cb * y4.y) / den,
                          (ca * h4.z + cb * y4.z) / den, (ca * h4.w + cb * y4.w) / den);
  float sn = sqrtf(wave_sum(dot4(s4, s4)));
  float ps3 = proj_scale(sn, kc);
  float snp = sn * ps3;

  // logmap0
  float snc = fmaxf(snp, EPSF);
  float lf = artan_k(snc, kc) / snc * ps3;
  float4 t4 = scl4(s4, lf);
  ((float4*)(mx + (size_t)w * DD))[lane] = t4;

  // fused self-loop init of aggregation buffer
  float dv = dinv[w];
  ((float4*)(agg + (size_t)w * DD))[lane] = scl4(t4, dv * dv);
}

// final expmap0(agg) -> logmap0 -> batch-mean accumulate into features
__global__ void k_expert_final(const float* __restrict__ agg, const int* __restrict__ batch,
                               const float* __restrict__ counts, float* __restrict__ feat,
                               float kc, int expert, int nrows) {
  int w = (blockIdx.x * blockDim.x + threadIdx.x) >> 5;
  int lane = threadIdx.x & 31;
  if (w >= nrows) return;
  float4 a4 = ((const float4*)(agg + (size_t)w * DD))[lane];
  float an = sqrtf(wave_sum(dot4(a4, a4)));
  float anc = fmaxf(an, EPSF);
  float f = tan_k(anc, kc) / anc;
  float xn = f * an;
  float ps = proj_scale(xn, kc);
  f *= ps; xn *= ps;
  float xnc = fmaxf(xn, EPSF);
  float lf = artan_k(xnc, kc) / xnc * f;

  int b = batch[w];
  float inv = 1.f / fmaxf(counts[b], 1.f);
  float4 z = scl4(a4, lf * inv);
  float* dst = feat + (size_t)b * (KEXP * DD) + expert * DD + lane * 4;
  atomicAdd(dst + 0, z.x);
  atomicAdd(dst + 1, z.y);
  atomicAdd(dst + 2, z.z);
  atomicAdd(dst + 3, z.w);
}

// ---------------- graph aggregation ----------------

__global__ void k_selfinit(const float* __restrict__ src, float* __restrict__ dst,
                           const float* __restrict__ dinv, int n, int C) {
  int i = blockIdx.x * blockDim.x + threadIdx.x;
  if (i >= n) return;
  float dv = dinv[i / C];
  dst[i] = dv * dv * src[i];
}

__global__ void k_scatter(const float* __restrict__ src, float* __restrict__ dst,
                          const int* __restrict__ row, const int* __restrict__ col,
                          const float* __restrict__ dinv, int E, int C) {
  int w = (blockIdx.x * blockDim.x + threadIdx.x) >> 5;
  int lane = threadIdx.x & 31;
  if (w >= E) return;
  if (w + 8 < E) {                                  // lookahead gather prefetch
    int rn = row[w + 8];
    __builtin_prefetch(src + (size_t)rn * C + (lane & (C - 1)), 0, 1);
  }
  int r = row[w], c = col[w];
  float wt = dinv[r] * dinv[c];
  const float* s = src + (size_t)r * C;
  float* d = dst + (size_t)c * C;
  for (int j = lane; j < C; j += 32) atomicAdd(d + j, wt * s[j]);
}

__global__ void k_bias_relu(float* __restrict__ buf, const float* __restrict__ b, int n, int C) {
  int i = blockIdx.x * blockDim.x + threadIdx.x;
  if (i >= n) return;
  buf[i] = fmaxf(buf[i] + b[i % C], 0.f);
}

__global__ void k_gate_pool(const float* __restrict__ g, const int* __restrict__ batch,
                            const float* __restrict__ counts, float* __restrict__ hgate, int nrows) {
  int i = blockIdx.x * blockDim.x + threadIdx.x;
  if (i >= nrows * DD) return;
  int n = i >> 7, c = i & 127;
  int b = batch[n];
  atomicAdd(&hgate[(size_t)b * DD + c], g[i] / fmaxf(counts[b], 1.f));
}

// ---------------- geometric distance gate (one wave per graph) ----------------

__global__ void k_gate_dist(const float* __restrict__ hgate, const float* __restrict__ gate_u,
                            const float* __restrict__ tau_raw, float* __restrict__ out) {
  int b = blockIdx.x;
  int lane = threadIdx.x & 31;
  const float curvs[KEXP] = {-1.f, 0.f, 1.f, -0.5f};

  float4 h4 = ((const float4*)(hgate + (size_t)b * DD))[lane];
  float hn = sqrtf(wave_sum(dot4(h4, h4)));

  float d[KEXP];
#pragma unroll
  for (int i = 0; i < KEXP; ++i) {
    float kc = curvs[i];
    float hnc = fmaxf(hn, EPSF);
    float zf = tan_k(hnc, kc) / hnc;
    float zn = zf * hn;
    float p = proj_scale(zn, kc);
    zf *= p; zn *= p;
    float4 z4 = scl4(h4, zf);

    float4 u4 = ((const float4*)(gate_u + (size_t)i * DD))[lane];
    float un = sqrtf(wave_sum(dot4(u4, u4)));
    float unc = fmaxf(un, EPSF);
    float yf = tan_k(unc, kc) / unc;
    float yn = yf * un;
    p = proj_scale(yn, kc);
    yf *= p; yn *= p;
    float4 y4 = scl4(u4, yf);

    // mobius_add(-z, y)
    float x2 = zn * zn, y2 = yn * yn;
    float xy = -wave_sum(dot4(z4, y4));
    float ca = 1.f - 2.f * kc * xy - kc * y2;
    float cb = 1.f + kc * x2;
    float den = fmaxf(1.f - 2.f * kc * xy + kc * kc * x2 * y2, 1e-15f);
    float4 m4 = make_float4((ca * -z4.x + cb * y4.x) / den, (ca * -z4.y + cb * y4.y) / den,
                            (ca * -z4.z + cb * y4.z) / den, (ca * -z4.w + cb * y4.w) / den);
    float mn = sqrtf(wave_sum(dot4(m4, m4)));
    mn *= proj_scale(mn, kc);
    d[i] = 2.f * artan_k(fmaxf(mn, EPSF), kc);
  }

  float tau[KEXP], lg[KEXP], wg[KEXP];
  float mx = -1e30f;
#pragma unroll
  for (int i = 0; i < KEXP; ++i) {
    float t = log1pf(expf(tau_raw[i])) + 0.05f;          // softplus + TAU_MIN
    tau[i] = fminf(fmaxf(t, 0.05f), 10.f);
    lg[i] = -d[i] / tau[i];
    mx = fmaxf(mx, lg[i]);
  }
  float se = 0.f;
#pragma unroll
  for (int i = 0; i < KEXP; ++i) { wg[i] = expf(lg[i] - mx); se += wg[i]; }

  if (lane == 0) {
#pragma unroll
    for (int i = 0; i < KEXP; ++i) {
      out[W_OFF + b * KEXP + i] = wg[i] / se;
      out[D_OFF + b * KEXP + i] = d[i];
    }
    if (b == 0) {
#pragma unroll
      for (int i = 0; i < KEXP; ++i) out[TAU_OFF + i] = tau[i];
    }
  }
}

__global__ void k_scale_features(float* __restrict__ out) {
  int i = blockIdx.x * blockDim.x + threadIdx.x;
  if (i >= W_OFF) return;
  int b = i >> 9;            // / 512
  int slot = (i >> 7) & 3;   // (% 512) / 128
  out[i] *= out[W_OFF + b * KEXP + slot];
}

// ---------------- host launcher ----------------

static inline int cdiv(int a, int b) { return (a + b - 1) / b; }

extern "C" void kernel_launch(void* const* d_in, const int* in_sizes, int n_in,
                              void* d_out, int out_size, void* d_ws, size_t ws_size,
                              hipStream_t stream) {
  const float* x       = (const float*)d_in[0];
  const int*   ei      = (const int*)d_in[1];
  const int*   batch   = (const int*)d_in[2];
  const float* ew1     = (const float*)d_in[3];
  const float* eb1     = (const float*)d_in[4];
  const float* ew2     = (const float*)d_in[5];
  const float* eb2     = (const float*)d_in[6];
  const float* gw1     = (const float*)d_in[7];
  const float* gb1     = (const float*)d_in[8];
  const float* gw2     = (const float*)d_in[9];
  const float* gb2     = (const float*)d_in[10];
  const float* gu      = (const float*)d_in[11];
  const float* tau_raw = (const float*)d_in[12];

  const int N = in_sizes[0] / DD;   // 50000
  const int E = in_sizes[1] / 2;    // 800000
  const int* row = ei;
  const int* col = ei + E;

  float* out = (float*)d_out;
  float* ws = (float*)d_ws;
  float* dinv   = ws;                                   // N   (deg then dinv)
  float* counts = dinv + N;                             // B
  float* hgate  = counts + BB;                          // B*128
  float* buf0   = hgate + (size_t)BB * DD;              // N*128
  float* buf1   = buf0 + (size_t)N * DD;                // N*128
  float* buf2   = buf1 + (size_t)N * DD;                // N*128

  const int BL = 256;

  // ---- init ----
  k_fill<<<cdiv(W_OFF, BL), BL, 0, stream>>>(out, 0.f, W_OFF);
  k_fill<<<1, BB, 0, stream>>>(counts, 0.f, BB);
  k_fill<<<cdiv(BB * DD, BL), BL, 0, stream>>>(hgate, 0.f, BB * DD);
  k_fill<<<cdiv(N, BL), BL, 0, stream>>>(dinv, 1.f, N);          // self-loop degree
  k_deg_accum<<<cdiv(E, BL), BL, 0, stream>>>(col, dinv, E);
  k_rsqrt_inplace<<<cdiv(N, BL), BL, 0, stream>>>(dinv, N);
  k_counts<<<cdiv(N, BL), BL, 0, stream>>>(batch, counts, N);

  // ---- Euclidean gate GCN ----
  k_gemm_lds<DD, GH><<<cdiv(N, 128), BL, 0, stream>>>(x, gw1, buf1, N);
  k_selfinit<<<cdiv(N * GH, BL), BL, 0, stream>>>(buf1, buf2, dinv, N * GH, GH);
  k_scatter<<<cdiv(E * 32, BL), BL, 0, stream>>>(buf1, buf2, row, col, dinv, E, GH);
  k_bias_relu<<<cdiv(N * GH, BL), BL, 0, stream>>>(buf2, gb1, N * GH, GH);
  k_gemm_lds<GH, DD><<<cdiv(N, 128), BL, 0, stream>>>(buf2, gw2, buf1, N);
  k_selfinit<<<cdiv(N * DD, BL), BL, 0, stream>>>(buf1, buf2, dinv, N * DD, DD);
  k_scatter<<<cdiv(E * 32, BL), BL, 0, stream>>>(buf1, buf2, row, col, dinv, E, DD);
  k_bias_relu<<<cdiv(N * DD, BL), BL, 0, stream>>>(buf2, gb2, N * DD, DD);
  k_gate_pool<<<cdiv(N * DD, BL), BL, 0, stream>>>(buf2, batch, counts, hgate, N);

  // ---- Riemannian experts ----
  const float curvs[KEXP] = {-1.f, 0.f, 1.f, -0.5f};
  for (int i = 0; i < KEXP; ++i) {
    float kc = curvs[i];
    const float* W1 = ew1 + (size_t)i * DD * DD;
    const float* b1 = eb1 + (size_t)i * DD;
    const float* W2 = ew2 + (size_t)i * DD * DD;
    const float* b2 = eb2 + (size_t)i * DD;

    k_expmap0<<<cdiv(N * 32, BL), BL, 0, stream>>>(x, buf0, kc, N);

    k_gemm_lds<DD, DD><<<cdiv(N, 128), BL, 0, stream>>>(buf0, W1, buf1, N);
    k_expert_post<<<cdiv(N * 32, BL), BL, 0, stream>>>(buf0, buf1, buf2, dinv, b1, kc, N);
    k_scatter<<<cdiv(E * 32, BL), BL, 0, stream>>>(buf1, buf2, row, col, dinv, E, DD);
    k_expmap0<<<cdiv(N * 32, BL), BL, 0, stream>>>(buf2, buf0, kc, N);

    k_gemm_lds<DD, DD><<<cdiv(N, 128), BL, 0, stream>>>(buf0, W2, buf1, N);
    k_expert_post<<<cdiv(N * 32, BL), BL, 0, stream>>>(buf0, buf1, buf2, dinv, b2, kc, N);
    k_scatter<<<cdiv(E * 32, BL), BL, 0, stream>>>(buf1, buf2, row, col, dinv, E, DD);

    k_expert_final<<<cdiv(N * 32, BL), BL, 0, stream>>>(buf2, batch, counts, out, kc, i, N);
  }

  // ---- distance gate + scaling ----
  k_gate_dist<<<BB, 32, 0, stream>>>(hgate, gu, tau_raw, out);
  k_scale_features<<<cdiv(W_OFF, BL), BL, 0, stream>>>(out);
}